// MnistSplineKAN_22771916603481
// MI455X (gfx1250) — compile-verified
//
#include <hip/hip_runtime.h>
#include <hip/hip_bf16.h>

// KAN forward as one fused f32 WMMA GEMM:
//   out[b,j] = sum_{i,kk} F[b,(i,kk)] * M'[(i,kk),j] + bias[j]
// F features per input i (kk=0..11): silu(x), 8 cubic B-spline values, 3 zero
// pads (K multiple of 4). M' folds mask*scale*coef AND the 10x10 output weight
// W (j padded 10->16). Contraction on V_WMMA_F32_16X16X4_F32 (full f32 math).
//
// Round-3 fix: silu's IEEE divide (v_div_scale/fmas/fixup chain) replaced by
// the hardware reciprocal v_rcp_f32 (1 ulp) -- a sigmoid doesn't need a
// correctly-rounded divide.

typedef float v2f __attribute__((ext_vector_type(2)));
typedef float v8f __attribute__((ext_vector_type(8)));

#define IN_DIM   784
#define GRID_PTS 12      // NUM+1 + 2*K = 6 + 6
#define NCHUNK   3       // 12 features / WMMA K=4
// d_ws layout: [0, 602112)  B-fragments: [IN_DIM][NCHUNK][32 lanes][2] floats
//              [602112, +6272) per-input grid params (g0, 1/h) as float2

// ---------------------------------------------------------------------------
// Kernel 0a: fold mask, scale_base, scale_sp, coef and W into WMMA B-fragments.
// float2-interleaved layout; element (i, c, lane, v) at ((i*3+c)*32+lane)*2+v:
//   .x (v=0): lanes 0-15 -> K=4c+0, lanes 16-31 -> K=4c+2
//   .y (v=1): lanes 0-15 -> K=4c+1, lanes 16-31 -> K=4c+3    (N = lane & 15)
// ---------------------------------------------------------------------------
__global__ void kan_fold_weights(const float* __restrict__ coef,
                                 const float* __restrict__ scale_base,
                                 const float* __restrict__ scale_sp,
                                 const float* __restrict__ mask,
                                 const float* __restrict__ W,
                                 float* __restrict__ Bfrag) {
    int t = blockIdx.x * blockDim.x + threadIdx.x;
    const int TOT = IN_DIM * NCHUNK * 2 * 32;
    if (t >= TOT) return;
    int i    = t / (NCHUNK * 64);
    int rem  = t % (NCHUNK * 64);
    int c    = rem / 64;
    int lane = (rem % 64) / 2;
    int v    = rem & 1;
    int kk   = 4 * c + v + 2 * (lane >> 4);   // feature index 0..11
    int j    = lane & 15;                     // output column 0..15
    float acc = 0.0f;
    if (j < 10 && kk < 9) {
        #pragma unroll
        for (int o = 0; o < 10; ++o) {
            float m = mask[i * 10 + o];
            float s = (kk == 0)
                        ? scale_base[i * 10 + o]
                        : scale_sp[i * 10 + o] * coef[(i * 10 + o) * 8 + (kk - 1)];
            acc += m * s * W[j * 10 + o];     // out = y @ W.T  ->  W[j,o]
        }
    }
    Bfrag[((i * NCHUNK + c) * 32 + lane) * 2 + v] = acc;
}

// ---------------------------------------------------------------------------
// Kernel 0b: per-input uniform-grid params (g0, 1/h).
// ---------------------------------------------------------------------------
__global__ void kan_grid_params(const float* __restrict__ grid,
                                float2* __restrict__ gp) {
    int i = blockIdx.x * blockDim.x + threadIdx.x;
    if (i >= IN_DIM) return;
    float g0 = grid[i * GRID_PTS];
    float h  = grid[i * GRID_PTS + 1] - g0;
    gp[i] = make_float2(g0, 1.0f / h);
}

// ---------------------------------------------------------------------------
// Main kernel: one wave (32 lanes) per 16-row batch tile. Each half-wave
// computes the full feature set for its row (lane & 15); the WMMA A-fragment
// (lanes 0-15: K={0,1}, lanes 16-31: K={2,3}) is assembled with 5 cndmasks.
// 3 chained f32 WMMAs per input i accumulate all 16 output columns.
// ---------------------------------------------------------------------------
__global__ __launch_bounds__(32)
void kan_main(const float* __restrict__ x,
              const float2* __restrict__ gp,
              const float2* __restrict__ Bfrag,
              const float* __restrict__ bias,
              float* __restrict__ out) {
    const int  lane = threadIdx.x;         // 0..31, full wave32
    const int  b0   = blockIdx.x * 16;     // batch tile base
    const int  row  = lane & 15;
    const bool hi   = lane >= 16;          // upper half-wave holds K={2,3}
    const float* xrow = x + (b0 + row) * IN_DIM;

    v8f acc = {0.f, 0.f, 0.f, 0.f, 0.f, 0.f, 0.f, 0.f};

    for (int i = 0; i < IN_DIM; ++i) {
        float  xv = xrow[i];
        float2 g  = gp[i];
        float  u  = (xv - g.x) * g.y;       // (x - g0) / h

        // silu via hardware reciprocal (v_rcp_f32, 1 ulp) instead of IEEE div
        float sil = xv * __builtin_amdgcn_rcpf(1.0f + __expf(-xv));

        // Degree-1 basis on a uniform grid is a hat function (exactly equal to
        // the reference Cox-de Boor recursion from half-open indicators):
        //   B1[t] = max(0, 1 - |u - (t+1)|)
        float B1[10], B2[9], B3[8];
        #pragma unroll
        for (int t = 0; t < 10; ++t)
            B1[t] = fmaxf(0.0f, 1.0f - fabsf(u - (float)(t + 1)));
        #pragma unroll
        for (int t = 0; t < 9; ++t)
            B2[t] = 0.5f * ((u - (float)t) * B1[t] + ((float)(t + 3) - u) * B1[t + 1]);
        #pragma unroll
        for (int t = 0; t < 8; ++t)
            B3[t] = (1.0f / 3.0f) *
                    ((u - (float)t) * B2[t] + ((float)(t + 4) - u) * B2[t + 1]);

        // A fragments via constant-index selects (feature map:
        // f0=silu, f1..f8=B3[0..7], f9..f11=pad with zero folded weights).
        v2f a0, a1, a2;
        a0.x = hi ? B3[1] : sil;     // K = 0 / 2
        a0.y = hi ? B3[2] : B3[0];   // K = 1 / 3
        a1.x = hi ? B3[5] : B3[3];   // K = 4 / 6
        a1.y = hi ? B3[6] : B3[4];   // K = 5 / 7
        a2.x = hi ? 0.0f  : B3[7];   // K = 8 / 10(pad)
        a2.y = 0.0f;                 // K = 9/11: pad, weights are zero

        const float2* bf = Bfrag + i * (NCHUNK * 32);
        float2 bv0 = bf[lane];            // one global_load_b64 per chunk
        float2 bv1 = bf[32 + lane];
        float2 bv2 = bf[64 + lane];
        v2f b0v; b0v.x = bv0.x; b0v.y = bv0.y;
        v2f b1v; b1v.x = bv1.x; b1v.y = bv1.y;
        v2f b2v; b2v.x = bv2.x; b2v.y = bv2.y;

        // D = A(16x4 f32) * B(4x16 f32) + C, full f32 accumulate
        acc = __builtin_amdgcn_wmma_f32_16x16x4_f32(
            false, a0, false, b0v, (short)0, acc, false, false);
        acc = __builtin_amdgcn_wmma_f32_16x16x4_f32(
            false, a1, false, b1v, (short)0, acc, false, false);
        acc = __builtin_amdgcn_wmma_f32_16x16x4_f32(
            false, a2, false, b2v, (short)0, acc, false, false);
    }

    // D layout: VGPR r -> (M = r + 8*(lane>=16), N = lane&15)
    int col = lane & 15;
    if (col < 10) {
        float bv = bias[col];
        #pragma unroll
        for (int r = 0; r < 8; ++r) {
            int m = r + (hi ? 8 : 0);
            out[(b0 + m) * 10 + col] = acc[r] + bv;
        }
    }
}

extern "C" void kernel_launch(void* const* d_in, const int* in_sizes, int n_in,
                              void* d_out, int out_size, void* d_ws, size_t ws_size,
                              hipStream_t stream) {
    const float* x          = (const float*)d_in[0];
    const float* grid       = (const float*)d_in[1];
    const float* coef       = (const float*)d_in[2];
    const float* scale_base = (const float*)d_in[3];
    const float* scale_sp   = (const float*)d_in[4];
    const float* mask       = (const float*)d_in[5];
    const float* W          = (const float*)d_in[6];
    const float* bias       = (const float*)d_in[7];
    float*       out        = (float*)d_out;

    float*  Bfrag = (float*)d_ws;                       // 602,112 B
    float2* gp    = (float2*)((char*)d_ws + IN_DIM * NCHUNK * 64 * 4);

    (void)in_sizes; (void)n_in; (void)out_size; (void)ws_size;

    const int tot = IN_DIM * NCHUNK * 2 * 32;
    kan_fold_weights<<<(tot + 255) / 256, 256, 0, stream>>>(
        coef, scale_base, scale_sp, mask, W, Bfrag);
    kan_grid_params<<<(IN_DIM + 255) / 256, 256, 0, stream>>>(grid, gp);

    // 8192 rows / 16-row tile = 512 tiles, one wave each
    kan_main<<<512, 32, 0, stream>>>(x, gp, (const float2*)Bfrag, bias, out);
}